// Temperature_Normalized_Contrastive_Loss_46265387713076
// MI455X (gfx1250) — compile-verified
//
#include <hip/hip_runtime.h>
#include <hip/hip_bf16.h>

// Temperature-normalized contrastive loss, MI455X (gfx1250, wave32).
//   loss = -(1/N) * sum_i [ s(i,jf_i) - logsumexp_{j!=i} s(i,j) ],  s = (A A^T)/T
//
// v4: as v3 (TDM-staged B tiles, double-buffered, 8 waves sharing LDS,
// 512 workgroups), but B fragments are bulk-loaded from LDS into registers
// before the WMMA chain so the DS pipe can run ahead of the matrix pipe
// (batch ds_load clause + partial s_wait_dscnt instead of dscnt==0 per pair).

#define NROWS  8192
#define DIM    128
#define INV_T  10.0f            // 1 / 0.1
#define NEG_BIG (-3.0e38f)
#define NCHUNK 8                // column chunks (blockIdx.y)
#define TILES_PER_CHUNK ((NROWS / 16) / NCHUNK)   // 64
#define LDS_STRIDE 132          // 128 + 4 dwords pad: stride % 64 == 4 -> conflict-free b64 reads

typedef __attribute__((ext_vector_type(2))) float    v2f;
typedef __attribute__((ext_vector_type(8))) float    v8f;
typedef __attribute__((ext_vector_type(4))) unsigned u32x4;
typedef __attribute__((ext_vector_type(8))) int      i32x8;
typedef __attribute__((ext_vector_type(4))) int      i32x4;

#if defined(__has_builtin)
#  if __has_builtin(__builtin_amdgcn_tensor_load_to_lds)
#    define USE_TDM 1
#  endif
#endif
#ifndef USE_TDM
#  define USE_TDM 0
#endif

__device__ __forceinline__ void wait_tensor0() {
#if defined(__has_builtin)
#  if __has_builtin(__builtin_amdgcn_s_wait_tensorcnt)
    __builtin_amdgcn_s_wait_tensorcnt(0);
    return;
#  else
    asm volatile("s_wait_tensorcnt 0x0" ::: "memory");
    return;
#  endif
#else
    asm volatile("s_wait_tensorcnt 0x0" ::: "memory");
#endif
}

#if USE_TDM
// One TDM transfer: 16x128 fp32 tile (rows colBase..+15 of A) -> LDS at
// lds_off, padded 4 dwords every 128 (LDS row stride = 132 dwords).
// D# per CDNA5 ISA ch.8: group0 {count=1, lds_addr, global_addr, type=2},
// group1 {data_size=4B, pad_enable, pad_interval=128dw, pad_amount=4dw,
//         tensor_dim0=128, tensor_dim1=8192, tile_dim0=128, tile_dim1=16,
//         tensor_dim0_stride=128}. Groups 2/3 unused (2D tensor).
__device__ __forceinline__ void tdm_load_tile(const float* gsrc, unsigned lds_off) {
    unsigned long long ga = (unsigned long long)(uintptr_t)gsrc;
    u32x4 g0;
    g0[0] = 1u;                                               // count=1 (valid user D#)
    g0[1] = lds_off;                                          // lds_addr (bytes)
    g0[2] = (unsigned)(ga & 0xffffffffull);                   // global_addr[31:0]
    g0[3] = (unsigned)((ga >> 32) & 0x01ffffffull) | (2u << 30); // addr[56:32], type=2
    i32x8 g1;
    g1[0] = (2 << 16) | (1 << 20) | (6 << 22) | (3 << 25);    // data_size=4B, pad: 4dw per 128dw
    g1[1] = (int)(128u << 16);                                // tensor_dim0 = 128   [79:48]
    g1[2] = (int)(8192u << 16);                               // tensor_dim1 = 8192  [111:80]
    g1[3] = (int)(128u << 16);                                // tile_dim0 = 128     [127:112]
    g1[4] = 16;                                               // tile_dim1 = 16, tile_dim2 = 0
    g1[5] = 128;                                              // tensor_dim0_stride = 128
    g1[6] = 0;
    g1[7] = 0;
    i32x4 z4 = {0, 0, 0, 0};
    i32x8 z8 = {0, 0, 0, 0, 0, 0, 0, 0};
    __builtin_amdgcn_tensor_load_to_lds(g0, g1, z4, z4, z8, 0);
}
#endif

// ---------------------------------------------------------------------------
// Kernel 1: per row, first j != i with target[j]==target[i]; s_first directly.
// ---------------------------------------------------------------------------
__global__ void tncl_first_pos(const float* __restrict__ A,
                               const int* __restrict__ target,
                               float* __restrict__ sfirst) {
    int i = blockIdx.x * blockDim.x + threadIdx.x;
    if (i >= NROWS) return;
    int tgt = target[i];
    int jf = -1;
    for (int j = 0; j < NROWS; ++j) {
        if (j != i && target[j] == tgt) { jf = j; break; }
    }
    float s = 0.0f;
    if (jf >= 0) {
        const float4* ai = (const float4*)(A + (size_t)i  * DIM);
        const float4* aj = (const float4*)(A + (size_t)jf * DIM);
        float acc = 0.0f;
#pragma unroll
        for (int k = 0; k < DIM / 4; ++k) {
            float4 x = ai[k], y = aj[k];
            acc += x.x * y.x + x.y * y.y + x.z * y.z + x.w * y.w;
        }
        s = acc * INV_T;
    }
    sfirst[i] = s;
}

// ---------------------------------------------------------------------------
// Kernel 2: grid (NROWS/128, NCHUNK), 256 threads = 8 waves. Wave w owns rows
// [blockIdx.x*128 + w*16, +16); A fragments register-resident. All waves sweep
// the chunk's 64 column tiles; each 16x128 B tile is TDM-DMA'd into LDS once
// (double-buffered) and consumed by all 8 waves: bulk ds_load of 32 B
// fragments, then 32x v_wmma_f32_16x16x4_f32, then online (max,sumexp) update.
//
// Fragment layout (ISA 7.12.2, 32-bit):
//   A 16x4 : lane L -> row L%16, K = 2*(L/16)+{0,1}  (contiguous float2)
//   B 4x16 : lane L -> col L%16, K = 2*(L/16)+{0,1}; B[k][n]=A[col][k]
//   C 16x16: VGPR v, lane L -> row v + 8*(L/16), col L%16
// LDS B reads: dword addr = l16*132 + 4kk + 2half -> banks {4*l16+2half} all
// distinct across the wave (stride 132 % 64 == 4).
// ---------------------------------------------------------------------------
__launch_bounds__(256)
__global__ void tncl_row_lse(const float* __restrict__ A,
                             float* __restrict__ pm,     // [NROWS][NCHUNK] partial max
                             float* __restrict__ pl) {   // [NROWS][NCHUNK] partial sumexp
    __shared__ float bbuf[2][16 * LDS_STRIDE];

    const int lane = threadIdx.x & 31;
    const int wave = threadIdx.x >> 5;
    const int half = lane >> 4;
    const int l16  = lane & 15;
    const int rowBaseW = blockIdx.x * 128 + wave * 16;
    const int tile0 = blockIdx.y * TILES_PER_CHUNK;

    // Preload this wave's 16x128 A fragments (32 K-steps of v2f per lane).
    const float2* A2 = (const float2*)(A + (size_t)(rowBaseW + l16) * DIM);
    v2f afrag[32];
#pragma unroll
    for (int kk = 0; kk < 32; ++kk) {
        float2 t = A2[2 * kk + half];
        afrag[kk] = (v2f){t.x, t.y};
    }

    float mx[8], ls[8];
#pragma unroll
    for (int v = 0; v < 8; ++v) { mx[v] = NEG_BIG; ls[v] = 0.0f; }

#if USE_TDM
    if (wave == 0)
        tdm_load_tile(A + (size_t)(tile0 * 16) * DIM,
                      (unsigned)(uintptr_t)&bbuf[0][0]);
#else
    {   // cooperative stage of tile0 into bbuf[0]
        const int colBase = tile0 * 16;
        for (int c = threadIdx.x; c < 512; c += 256) {
            int row = c >> 5, cw = c & 31;
            float4 vdat = *(const float4*)(A + (size_t)(colBase + row) * DIM + cw * 4);
            *(float4*)(&bbuf[0][row * LDS_STRIDE + cw * 4]) = vdat;
        }
    }
#endif

    for (int tt = 0; tt < TILES_PER_CHUNK; ++tt) {
        const int t = tile0 + tt;
        const int p = tt & 1;
#if USE_TDM
        if (wave == 0) wait_tensor0();
#endif
        __syncthreads();                       // bbuf[p] ready for all waves

        if (tt + 1 < TILES_PER_CHUNK) {        // prefetch next tile
#if USE_TDM
            if (wave == 0)
                tdm_load_tile(A + (size_t)((t + 1) * 16) * DIM,
                              (unsigned)(uintptr_t)&bbuf[p ^ 1][0]);
#else
            const int colBase = (t + 1) * 16;
            for (int c = threadIdx.x; c < 512; c += 256) {
                int row = c >> 5, cw = c & 31;
                float4 vdat = *(const float4*)(A + (size_t)(colBase + row) * DIM + cw * 4);
                *(float4*)(&bbuf[p ^ 1][row * LDS_STRIDE + cw * 4]) = vdat;
            }
#endif
        }

        // Bulk-load all 32 B fragments first (pipelined DS clause), then run
        // the WMMA chain back-to-back against register-resident operands.
        const float* bp = &bbuf[p][0];
        const int boff = l16 * LDS_STRIDE + 2 * half;
        v2f bfrag[32];
#pragma unroll
        for (int kk = 0; kk < 32; ++kk) {
            float2 tb = *(const float2*)(bp + boff + 4 * kk);
            bfrag[kk] = (v2f){tb.x, tb.y};
        }

        v8f acc = {0.f, 0.f, 0.f, 0.f, 0.f, 0.f, 0.f, 0.f};
#pragma unroll
        for (int kk = 0; kk < 32; ++kk) {
            acc = __builtin_amdgcn_wmma_f32_16x16x4_f32(
                false, afrag[kk], false, bfrag[kk], (short)0, acc, false, false);
        }

        const int gj = t * 16 + l16;
#pragma unroll
        for (int v = 0; v < 8; ++v) {
            float s = acc[v] * INV_T;
            const int gi = rowBaseW + v + half * 8;
            if (gi == gj) s = -__builtin_inff();       // mask diagonal
            float nm = fmaxf(mx[v], s);
            ls[v] = ls[v] * __expf(mx[v] - nm) + __expf(s - nm);
            mx[v] = nm;
        }
    }

    // Reduce across the 16 lanes sharing each row (xor masks stay in-half).
#pragma unroll
    for (int v = 0; v < 8; ++v) {
#pragma unroll
        for (int off = 1; off < 16; off <<= 1) {
            float om = __shfl_xor(mx[v], off, 32);
            float ol = __shfl_xor(ls[v], off, 32);
            float nm = fmaxf(mx[v], om);
            ls[v] = ls[v] * __expf(mx[v] - nm) + ol * __expf(om - nm);
            mx[v] = nm;
        }
    }
    if (l16 == 0) {
#pragma unroll
        for (int v = 0; v < 8; ++v) {
            int gi = rowBaseW + v + half * 8;
            pm[gi * NCHUNK + blockIdx.y] = mx[v];
            pl[gi * NCHUNK + blockIdx.y] = ls[v];
        }
    }
}

// ---------------------------------------------------------------------------
// Kernel 3: combine the NCHUNK (m,l) partials per row, form the per-row loss
// term, and tree-reduce in one block (fixed order -> deterministic).
// ---------------------------------------------------------------------------
__global__ void tncl_reduce(const float* __restrict__ sfirst,
                            const float* __restrict__ pm,
                            const float* __restrict__ pl,
                            float* __restrict__ out) {
    __shared__ float sm[256];
    float acc = 0.0f;
    for (int r = threadIdx.x; r < NROWS; r += 256) {
        float m = NEG_BIG, l = 0.0f;
#pragma unroll
        for (int c = 0; c < NCHUNK; ++c) {
            float om = pm[r * NCHUNK + c], ol = pl[r * NCHUNK + c];
            float nm = fmaxf(m, om);
            l = l * __expf(m - nm) + ol * __expf(om - nm);
            m = nm;
        }
        acc += -(sfirst[r] - m - __logf(l)) * (1.0f / NROWS);
    }
    sm[threadIdx.x] = acc;
    __syncthreads();
    for (int s = 128; s > 0; s >>= 1) {
        if ((int)threadIdx.x < s) sm[threadIdx.x] += sm[threadIdx.x + s];
        __syncthreads();
    }
    if (threadIdx.x == 0) out[0] = sm[0];
}

extern "C" void kernel_launch(void* const* d_in, const int* in_sizes, int n_in,
                              void* d_out, int out_size, void* d_ws, size_t ws_size,
                              hipStream_t stream) {
    const float* A      = (const float*)d_in[0];
    const int*   target = (const int*)d_in[1];
    float* ws      = (float*)d_ws;
    float* sfirst  = ws;                          // NROWS
    float* pm      = ws + NROWS;                  // NROWS * NCHUNK
    float* pl      = pm + (size_t)NROWS * NCHUNK; // NROWS * NCHUNK

    tncl_first_pos<<<NROWS / 256, 256, 0, stream>>>(A, target, sfirst);
    dim3 grid(NROWS / 128, NCHUNK);
    tncl_row_lse<<<grid, 256, 0, stream>>>(A, pm, pl);
    tncl_reduce<<<1, 256, 0, stream>>>(sfirst, pm, pl, (float*)d_out);
}